// AttentionDownsampler_40561671143506
// MI455X (gfx1250) — compile-verified
//
#include <hip/hip_runtime.h>
#include <hip/hip_bf16.h>
#include <math.h>

// ---------------------------------------------------------------------------
// AttentionDownsampler fused kernel for MI455X (gfx1250, wave32, WMMA).
//
//   hr_feats   (4,256,224,224) f32
//   attn_kernel(9,256,3,3)     f32
//   kernel_bias(9,)            f32
//   logit_bias (1,9,1,1)       f32
//   keep_mask  (4,9,112,112)   bool (1 byte/elem assumed)
//   out        (4,256,112,112) f32
//
// Per (b, output row h) workgroup (224 threads = 7 wave32):
//   pass 1: per 8-channel chunk: separable 5x5 blur, horizontal pass scatters
//           straight into an im2col patch buffer patch[k][w]; then
//           V_WMMA_F32_16X16X4_F32 accumulates the 9x2304 logits GEMM
//           (M=9 padded to 16, N=16 pixels/wave, K stepped 4 at a time with
//           affine LDS addressing).
//   mask/bias + softmax over the 9 window positions.
//   pass 2: recompute blur per chunk, 9-tap dot of patch rows with attn.
// ---------------------------------------------------------------------------

typedef float v2f __attribute__((ext_vector_type(2)));
typedef float v8f __attribute__((ext_vector_type(8)));

#define B_      4
#define C_      256
#define H_      224
#define F_      112
#define KK_     9
#define CC_     8                   // channels per LDS chunk
#define NCHUNK  (C_ / CC_)          // 32
#define KC_     (CC_ * KK_)         // 72  (K elements per chunk)
#define STEPS_  (KC_ / 4)           // 18  (WMMA steps per chunk)
#define TSTEPS  (NCHUNK * STEPS_)   // 576 (total WMMA steps, K = 2304)

#define VBW 232                     // vertical-blur row stride (x in -3..226)
#define PW  F_                      // patch row stride (one slot per pixel)
#define LGW 16                      // logit row stride (9 padded to 16)

// separable gaussian g = f/sum(f), f = exp(d^2 * -sqrt(2)), d=-2..2
__device__ __constant__ float G5[5] = {
    0.00233959f, 0.16281372f, 0.66969341f, 0.16281372f, 0.00233959f};

// ---------------------------------------------------------------------------
// Kernel 0: pre-swizzle attn_kernel into per-step A-fragments matching the
// CDNA5 32-bit A 16x4 VGPR layout:
//   lane L: M = L%16 ; lanes 0-15 hold (K=kb+0, kb+1), lanes 16-31 (kb+2,kb+3)
// Rows M=9..15 are zero (padding of the 9 output channels to 16).
// K order = attn_kernel's natural (c, kh, kw) flattening -> c*9 + kh*3 + kw.
// ---------------------------------------------------------------------------
__global__ void attn_wswizzle(const float* __restrict__ wk,
                              float2* __restrict__ wfrag) {
    int idx  = blockIdx.x * blockDim.x + threadIdx.x;   // 0 .. 576*32-1
    int s    = idx >> 5;
    int lane = idx & 31;
    int kb   = s * 4 + ((lane & 16) ? 2 : 0);
    int m    = lane & 15;
    float2 v = make_float2(0.f, 0.f);
    if (m < KK_) {
        v.x = wk[m * (C_ * KK_) + kb];
        v.y = wk[m * (C_ * KK_) + kb + 1];
    }
    wfrag[idx] = v;
}

// ---------------------------------------------------------------------------
// Fused main kernel: one workgroup per (b, output row h).
// ---------------------------------------------------------------------------
__global__ __launch_bounds__(224)
void attn_downsample(const float* __restrict__ hr,
                     const float* __restrict__ wfrag,
                     const float* __restrict__ kbias,
                     const float* __restrict__ lbias,
                     const unsigned char* __restrict__ mask,
                     float* __restrict__ out) {
    const int b    = blockIdx.x / F_;
    const int h    = blockIdx.x % F_;
    const int tid  = threadIdx.x;
    const int lane = tid & 31;
    const int wav  = tid >> 5;                 // 0..6

    __shared__ float vb_s[CC_ * 3 * VBW];      // vertical-blur rows
    __shared__ float patch_s[KC_ * PW];        // im2col patches: [k_local][w]
    __shared__ float lg_s[F_ * LGW];           // logits -> attn (stride 16)

    const float g0 = G5[0], g1 = G5[1], g2 = G5[2], g3 = G5[3], g4 = G5[4];

    // --- separable 5x5 blur of channels [c0,c0+CC_), rows 2h-1..2h+1,
    //     horizontal pass scatters directly into patch_s[k_local][w] with
    //     k_local = cl*9 + r*3 + kw and patch value = blur[r][2w+kw-1]. ---
    auto do_blur = [&](int c0) {
        // vertical: vb[r][x] = sum_dy hr[2h-1+r+dy][x]*g[dy+2]
        for (int task = tid; task < CC_ * 230; task += 224) {
            int cl = task / 230;
            int x  = task % 230 - 3;                    // -3 .. 226
            const int cbase = (b * C_ + c0 + cl) * H_;
            float r[7];
#pragma unroll
            for (int i = 0; i < 7; ++i) {
                int y = 2 * h - 3 + i;
                r[i] = (x >= 0 && x < H_ && y >= 0 && y < H_)
                           ? hr[(cbase + y) * H_ + x] : 0.f;
            }
#pragma unroll
            for (int rr = 0; rr < 3; ++rr) {
                float a = r[rr] * g0;
                a = fmaf(r[rr + 1], g1, a);
                a = fmaf(r[rr + 2], g2, a);
                a = fmaf(r[rr + 3], g3, a);
                a = fmaf(r[rr + 4], g4, a);
                vb_s[(cl * 3 + rr) * VBW + (x + 3)] = a;
            }
        }
        __syncthreads();
        // horizontal + im2col scatter; blurred x in -1..223
        for (int task = tid; task < CC_ * 225; task += 224) {
            int cl = task / 225;
            int x  = task % 225 - 1;                    // -1 .. 223
            float a3[3];
#pragma unroll
            for (int rr = 0; rr < 3; ++rr) {
                const float* vrow = &vb_s[(cl * 3 + rr) * VBW + (x + 1)]; // x-2+3
                float a = vrow[0] * g0;
                a = fmaf(vrow[1], g1, a);
                a = fmaf(vrow[2], g2, a);
                a = fmaf(vrow[3], g3, a);
                a = fmaf(vrow[4], g4, a);
                a3[rr] = a;
            }
            int kb9 = cl * KK_;
            if (x & 1) {                       // x odd -> kw = 0 (w=(x+1)/2), kw = 2 (w=(x-1)/2)
                int w0 = (x + 1) >> 1;         // 0..112
                if (w0 < F_) {
#pragma unroll
                    for (int rr = 0; rr < 3; ++rr)
                        patch_s[(kb9 + rr * 3 + 0) * PW + w0] = a3[rr];
                }
                if (w0 > 0) {
#pragma unroll
                    for (int rr = 0; rr < 3; ++rr)
                        patch_s[(kb9 + rr * 3 + 2) * PW + (w0 - 1)] = a3[rr];
                }
            } else {                           // x even -> kw = 1 at w = x/2
                int w = x >> 1;                // 0..111
#pragma unroll
                for (int rr = 0; rr < 3; ++rr)
                    patch_s[(kb9 + rr * 3 + 1) * PW + w] = a3[rr];
            }
        }
        __syncthreads();
    };

    // ---------------- pass 1: logits via WMMA ----------------
    v8f acc = {0.f, 0.f, 0.f, 0.f, 0.f, 0.f, 0.f, 0.f};
    const int pw   = wav * 16 + (lane & 15);   // this lane's pixel (N index)
    const int offh = (lane & 16) ? 2 : 0;      // K pair selected by lane half

    for (int ch = 0; ch < NCHUNK; ++ch) {
        // warm next chunk's first source rows while this one is processed
        if (ch + 1 < NCHUNK && tid < CC_) {
            int y = 2 * h - 3; if (y < 0) y = 0;
            __builtin_prefetch(&hr[((b * C_ + (ch + 1) * CC_ + tid) * H_ + y) * H_], 0, 0);
        }
        do_blur(ch * CC_);

        const v2f* wf = (const v2f*)wfrag + (size_t)(ch * STEPS_) * 32;
        const float* bp = &patch_s[offh * PW + pw];   // affine B gather base
        for (int s = 0; s < STEPS_; ++s) {
            v2f a = wf[s * 32 + lane];                // pre-swizzled A fragment
            v2f bv;
            bv.x = bp[0];                             // B[k = 4s+off  ][pw]
            bv.y = bp[PW];                            // B[k = 4s+off+1][pw]
            acc = __builtin_amdgcn_wmma_f32_16x16x4_f32(
                false, a, false, bv, (short)0, acc, false, false);
            bp += 4 * PW;
        }
        __syncthreads();   // patch_s consumed; safe to overwrite next chunk
    }

    // scatter logits per C/D layout: VGPR v -> M = v + (lane<16 ? 0 : 8).
    // Row padded to 16 so all 8 VGPRs store unconditionally & contiguously.
    {
        const int mbase = (lane & 16) ? 8 : 0;
        float* lrow = &lg_s[pw * LGW + mbase];
#pragma unroll
        for (int v = 0; v < 8; ++v) lrow[v] = acc[v];
    }
    __syncthreads();

    // ---------------- mask + biases + softmax over 9 positions ----------------
    if (tid < F_) {
        int w = tid;
        float l[KK_];
        float mx = -1e30f;
#pragma unroll
        for (int k = 0; k < KK_; ++k) {
            float mf = mask[((b * KK_ + k) * F_ + h) * F_ + w] ? 1.f : 0.f;
            float lv = (lg_s[w * LGW + k] + kbias[k]) * mf + lbias[k];
            l[k] = lv;
            mx = fmaxf(mx, lv);
        }
        float sum = 0.f;
#pragma unroll
        for (int k = 0; k < KK_; ++k) { l[k] = __expf(l[k] - mx); sum += l[k]; }
        float inv = 1.f / sum;
#pragma unroll
        for (int k = 0; k < KK_; ++k) lg_s[w * LGW + k] = l[k] * inv;
    }
    __syncthreads();

    // ---------------- pass 2: 9-tap weighted sum from patch rows ----------------
    for (int ch = 0; ch < NCHUNK; ++ch) {
        do_blur(ch * CC_);
        for (int task = tid; task < CC_ * F_; task += 224) {   // exactly 4 iters
            int cl = task / F_;
            int w  = task - cl * F_;
            float o = 0.f;
#pragma unroll
            for (int kk = 0; kk < KK_; ++kk)
                o = fmaf(patch_s[(cl * KK_ + kk) * PW + w],
                         lg_s[w * LGW + kk], o);
            out[((size_t)(b * C_ + ch * CC_ + cl) * F_ + h) * F_ + w] = o;
        }
        __syncthreads();
    }
}

// ---------------------------------------------------------------------------
extern "C" void kernel_launch(void* const* d_in, const int* in_sizes, int n_in,
                              void* d_out, int out_size, void* d_ws, size_t ws_size,
                              hipStream_t stream) {
    const float*         hr  = (const float*)d_in[0];          // hr_feats
    const float*         wk  = (const float*)d_in[1];          // attn_kernel
    const float*         kb  = (const float*)d_in[2];          // kernel_bias
    const float*         lb  = (const float*)d_in[3];          // logit_bias
    const unsigned char* km  = (const unsigned char*)d_in[4];  // keep_mask (bool)
    float*               out = (float*)d_out;

    // workspace: pre-swizzled A-fragments, 576 steps * 32 lanes * float2 = 144 KiB
    float* wfrag = (float*)d_ws;

    attn_wswizzle<<<(TSTEPS * 32) / 256, 256, 0, stream>>>(wk, (float2*)wfrag);
    attn_downsample<<<B_ * F_, 224, 0, stream>>>(hr, wfrag, kb, lb, km, out);
}